// ImprovedRoIHeads_45698452030234
// MI455X (gfx1250) — compile-verified
//
#include <hip/hip_runtime.h>
#include <math.h>

typedef __attribute__((ext_vector_type(2))) float v2f;
typedef __attribute__((ext_vector_type(8))) float v8f;

#define BN      16000     // B*P rows
#define FDIM    1024
#define NC      91
#define NCAT    512       // padded combined output width
#define REG_OFF 128       // regression columns start
#define NIMG    16
#define PPI     1000
#define MM      90000     // P*(C-1) candidates per image
#define NDETS   100
#define OFFU    802.0f    // max(H,W)+2
#define SCORE_T 0.05f
#define NMS_T   0.5f
#define MIN_SZ  1e-2f
#define WCLIP   4.135166556742356f   // log(1000/16)
#define IMW     800.0f
#define IMH     800.0f

// ---------------- kernel 1: pack weights/bias, K-pair interleaved -----------------------
// Wcat2 layout: element (k, n) stored at [((k>>1)*NCAT + n)*2 + (k&1)], i.e. rows k, k+1
// of the combined weight matrix are interleaved as float2 so the WMMA B-fragment
// (.x = row k, .y = row k+1, same column) is one aligned b64 load.
__global__ void pack_w(const float* __restrict__ Wc, const float* __restrict__ bc,
                       const float* __restrict__ Wb, const float* __restrict__ bb,
                       float* __restrict__ Wcat2, float* __restrict__ bcat) {
    int t = blockIdx.x * blockDim.x + threadIdx.x;
    if (t < FDIM * NCAT) {
        int k = t >> 9;          // /512
        int n = t & (NCAT - 1);  // %512
        float v = 0.0f;
        if (n < NC)                               v = Wc[k * NC + n];
        else if (n >= REG_OFF && n < REG_OFF+364) v = Wb[k * 364 + (n - REG_OFF)];
        Wcat2[(((size_t)(k >> 1) * NCAT + n) << 1) + (k & 1)] = v;
    }
    if (t < NCAT) {
        float v = 0.0f;
        if (t < NC)                               v = bc[t];
        else if (t >= REG_OFF && t < REG_OFF+364) v = bb[t - REG_OFF];
        bcat[t] = v;
    }
}

// ---------------- kernel 2: fp32 WMMA GEMM  Y[16000x512] = X[16000x1024] @ Wcat + bcat ---
// 128 threads = 4 waves; each wave computes one 16(M) x 64(N) tile (4 WMMA N-tiles).
// A fragment: lane l holds X[m0+(l&15)][k + 2*(l>>4) + {0,1}]  (one b64 load)
// B fragment: lane l holds Wcat[k + 2*(l>>4) + {0,1}][n]       (one b64 load, packed layout)
__global__ __launch_bounds__(128) void gemm_wmma(const float* __restrict__ X,
                                                 const float* __restrict__ Wt2,
                                                 const float* __restrict__ bcat,
                                                 float* __restrict__ Y) {
    const int lane = threadIdx.x & 31;
    const int wav  = threadIdx.x >> 5;
    const int m0   = (blockIdx.y * 4 + wav) * 16;   // gridDim.y = 250
    const int n0   = blockIdx.x * 64;               // gridDim.x = 8
    const int lm   = lane & 15;
    const int lh   = lane >> 4;

    const float* __restrict__ arow = X + (size_t)(m0 + lm) * FDIM + 2 * lh;
    // kpair base for this lane: row pair index = k/2 + lh
    const float* __restrict__ bbase = Wt2 + (((size_t)lh * NCAT + n0 + lm) << 1);

    v8f acc0 = {}, acc1 = {}, acc2 = {}, acc3 = {};

    #pragma unroll 4
    for (int k = 0; k < FDIM; k += 4) {
        v2f a;
        a.x = arow[k];
        a.y = arow[k + 1];
        const v2f* __restrict__ bp = (const v2f*)(bbase + (size_t)(k >> 1) * (NCAT * 2));
        v2f b0 = bp[0];
        v2f b1 = bp[16];
        v2f b2 = bp[32];
        v2f b3 = bp[48];
        acc0 = __builtin_amdgcn_wmma_f32_16x16x4_f32(false, a, false, b0, (short)0, acc0, false, false);
        acc1 = __builtin_amdgcn_wmma_f32_16x16x4_f32(false, a, false, b1, (short)0, acc1, false, false);
        acc2 = __builtin_amdgcn_wmma_f32_16x16x4_f32(false, a, false, b2, (short)0, acc2, false, false);
        acc3 = __builtin_amdgcn_wmma_f32_16x16x4_f32(false, a, false, b3, (short)0, acc3, false, false);
    }

    const int n = n0 + lm;
    #pragma unroll
    for (int i = 0; i < 8; ++i) {
        int m = m0 + i + 8 * lh;
        float* yr = Y + (size_t)m * NCAT + n;
        yr[0]  = acc0[i] + bcat[n + 0];
        yr[16] = acc1[i] + bcat[n + 16];
        yr[32] = acc2[i] + bcat[n + 32];
        yr[48] = acc3[i] + bcat[n + 48];
    }
}

// ---------------- kernel 3: softmax + box decode/clip + filter ----------------
__global__ void decode_k(const float* __restrict__ Y, const float* __restrict__ props,
                         float* __restrict__ boxes_off, float* __restrict__ scores_m) {
    int idx = blockIdx.x * blockDim.x + threadIdx.x;
    if (idx >= BN) return;
    int b = idx / PPI, p = idx % PPI;
    const float* row = Y + (size_t)idx * NCAT;

    float mx = row[0];
    for (int c = 1; c < NC; ++c) mx = fmaxf(mx, row[c]);
    float sum = 0.0f;
    for (int c = 0; c < NC; ++c) sum += expf(row[c] - mx);
    float inv = 1.0f / sum;

    const float* pr = props + (size_t)idx * 4;
    float px1 = pr[0], py1 = pr[1], px2 = pr[2], py2 = pr[3];
    float w  = px2 - px1, h = py2 - py1;
    float cx = px1 + 0.5f * w, cy = py1 + 0.5f * h;

    float* bo = boxes_off + ((size_t)b * MM + (size_t)p * 90) * 4;
    float* sm = scores_m  +  (size_t)b * MM + (size_t)p * 90;

    for (int c = 1; c < NC; ++c) {
        const float* rg = row + REG_OFF + c * 4;
        float dx = rg[0], dy = rg[1];
        float dw = fminf(rg[2], WCLIP), dh = fminf(rg[3], WCLIP);
        float pcx = dx * w + cx, pcy = dy * h + cy;
        float pw = expf(dw) * w, ph = expf(dh) * h;
        float bx1 = fminf(fmaxf(pcx - 0.5f * pw, 0.0f), IMW);
        float by1 = fminf(fmaxf(pcy - 0.5f * ph, 0.0f), IMH);
        float bx2 = fminf(fmaxf(pcx + 0.5f * pw, 0.0f), IMW);
        float by2 = fminf(fmaxf(pcy + 0.5f * ph, 0.0f), IMH);
        float sc  = expf(row[c] - mx) * inv;
        bool keep = (sc > SCORE_T) && ((bx2 - bx1) >= MIN_SZ) && ((by2 - by1) >= MIN_SZ);
        float off = (float)c * OFFU;
        int ml = c - 1;
        bo[ml * 4 + 0] = bx1 + off;
        bo[ml * 4 + 1] = by1 + off;
        bo[ml * 4 + 2] = bx2 + off;
        bo[ml * 4 + 3] = by2 + off;
        sm[ml] = keep ? sc : -INFINITY;
    }
}

// ---------------- kernel 4: per-image greedy class-aware NMS + output gather -------------
#define NMS_NT 1024
__global__ __launch_bounds__(NMS_NT) void nms_k(const float* __restrict__ boxes_off,
                                                float* __restrict__ scores,
                                                float* __restrict__ out) {
    const int b   = blockIdx.x;
    const int tid = threadIdx.x;
    __shared__ float sv[NMS_NT];
    __shared__ int   si[NMS_NT];

    const float* bx = boxes_off + (size_t)b * MM * 4;
    float*       sc = scores    + (size_t)b * MM;

    for (int it = 0; it < NDETS; ++it) {
        // --- block-wide argmax with first-index tie-break (matches jnp.argmax) ---
        float best = -INFINITY;
        int   bi   = 0x7FFFFFFF;
        for (int m = tid; m < MM; m += NMS_NT) {
            float v = sc[m];
            if (v > best || (v == best && m < bi)) { best = v; bi = m; }
        }
        sv[tid] = best; si[tid] = bi;
        __syncthreads();
        for (int s = NMS_NT / 2; s > 0; s >>= 1) {
            if (tid < s) {
                float ov = sv[tid + s]; int oi = si[tid + s];
                if (ov > sv[tid] || (ov == sv[tid] && oi < si[tid])) { sv[tid] = ov; si[tid] = oi; }
            }
            __syncthreads();
        }
        int   j  = si[0];
        float s0 = sv[0];
        __syncthreads();

        float jx1 = bx[j * 4 + 0], jy1 = bx[j * 4 + 1];
        float jx2 = bx[j * 4 + 2], jy2 = bx[j * 4 + 3];
        float ja  = (jx2 - jx1) * (jy2 - jy1);

        // --- suppression pass (suppresses j itself too, like the reference) ---
        for (int m = tid; m < MM; m += NMS_NT) {
            float x1 = bx[m * 4 + 0], y1 = bx[m * 4 + 1];
            float x2 = bx[m * 4 + 2], y2 = bx[m * 4 + 3];
            float ix1 = fmaxf(jx1, x1), iy1 = fmaxf(jy1, y1);
            float ix2 = fminf(jx2, x2), iy2 = fminf(jy2, y2);
            float inter = fmaxf(ix2 - ix1, 0.0f) * fmaxf(iy2 - iy1, 0.0f);
            float a2 = (x2 - x1) * (y2 - y1);
            float iou = inter / (ja + a2 - inter + 1e-9f);
            if (iou > NMS_T) sc[m] = -INFINITY;
        }

        if (tid == 0) {
            bool  valid = s0 > 0.0f;
            int   lab   = valid ? (j % 90) + 1 : 0;
            float offv  = (float)lab * OFFU;
            float* ob = out + ((size_t)b * NDETS + it) * 4;
            if (valid) {
                ob[0] = jx1 - offv; ob[1] = jy1 - offv;
                ob[2] = jx2 - offv; ob[3] = jy2 - offv;
            } else {
                ob[0] = 0.0f; ob[1] = 0.0f; ob[2] = 0.0f; ob[3] = 0.0f;
            }
            out[NIMG * NDETS * 4 + b * NDETS + it]                = valid ? s0 : 0.0f;
            out[NIMG * NDETS * 4 + NIMG * NDETS + b * NDETS + it] = (float)lab;
        }
        __syncthreads();
    }
}

extern "C" void kernel_launch(void* const* d_in, const int* in_sizes, int n_in,
                              void* d_out, int out_size, void* d_ws, size_t ws_size,
                              hipStream_t stream) {
    const float* X     = (const float*)d_in[0];  // box_features [16000,1024]
    const float* Wc    = (const float*)d_in[1];  // [1024,91]
    const float* bc    = (const float*)d_in[2];  // [91]
    const float* Wb    = (const float*)d_in[3];  // [1024,364]
    const float* bb    = (const float*)d_in[4];  // [364]
    const float* props = (const float*)d_in[5];  // [16,1000,4]
    float* out = (float*)d_out;

    float* ws = (float*)d_ws;
    float* Wcat2     = ws;                                  // 1024*512 = 524288
    float* bcat      = Wcat2 + (size_t)FDIM * NCAT;         // 512
    float* Y         = ws + 524800;                         // 16000*512 = 8192000
    float* boxes_off = Y + (size_t)BN * NCAT;               // 16*90000*4 = 5760000
    float* scores_m  = boxes_off + (size_t)NIMG * MM * 4;   // 16*90000 = 1440000
    (void)ws_size; (void)n_in; (void)in_sizes; (void)out_size;

    // 1) pack weights (K-pair interleaved for b64 B-fragment loads)
    {
        int total = FDIM * NCAT;
        pack_w<<<(total + 255) / 256, 256, 0, stream>>>(Wc, bc, Wb, bb, Wcat2, bcat);
    }
    // 2) WMMA GEMM: grid (Ngroups=8, Mgroups=250), 4 waves/block
    gemm_wmma<<<dim3(8, 250), 128, 0, stream>>>(X, Wcat2, bcat, Y);
    // 3) decode + softmax + filter
    decode_k<<<(BN + 255) / 256, 256, 0, stream>>>(Y, props, boxes_off, scores_m);
    // 4) per-image NMS + outputs
    nms_k<<<NIMG, NMS_NT, 0, stream>>>(boxes_off, scores_m, out);
}